// RadialDCLayer_9122510536844
// MI455X (gfx1250) — compile-verified
//
#include <hip/hip_runtime.h>
#include <math.h>

// ---------------- static config (matches reference) ----------------
#define H_     256
#define W_     256
#define T_     24
#define G_     512            // GX = GY = 2*W
#define NRAD_  24576          // samples per frame
#define JW     6              // KB kernel width
#define ALPHA_ 14.04f         // 2.34 * J
#define TWO_PI_F 6.283185307179586476925f
#define KC_    64             // K-chunk staged in LDS per iteration

typedef __attribute__((ext_vector_type(2))) float v2f;
typedef __attribute__((ext_vector_type(8))) float v8f;

// ---------------- device helpers ----------------
__device__ __forceinline__ float bessel_i0f(float x) {
    float ax = fabsf(x);
    if (ax < 3.75f) {
        float t = ax / 3.75f; t *= t;
        return 1.0f + t*(3.5156229f + t*(3.0899424f + t*(1.2067492f +
               t*(0.2659732f + t*(0.0360768f + t*0.0045813f)))));
    } else {
        float t = 3.75f / ax;
        float p = 0.39894228f + t*(0.01328592f + t*(0.00225319f + t*(-0.00157565f +
                  t*(0.00916281f + t*(-0.02057706f + t*(0.02635537f +
                  t*(-0.01647633f + t*0.00392377f)))))));
        return expf(ax) * rsqrtf(ax) * p;
    }
}

// Kaiser-Bessel kernel kb(u) = I0(alpha*sqrt(1-(2u/J)^2))/J, |u|<=J/2
__device__ __forceinline__ float kb_dev(float u) {
    float r = u * (2.0f / (float)JW);
    float z = 1.0f - r * r;
    z = z < 0.0f ? 0.0f : z;
    return bessel_i0f(ALPHA_ * sqrtf(z)) * (1.0f / (float)JW);
}

// ---------------- setup kernels ----------------
// 1D apodization coefficients: s1[i] = 1 / DTFT_kb((i - N/2)/G), N=256, G=512
__global__ void scale1d_kernel(float* __restrict__ s1) {
    int i = threadIdx.x;             // 0..255, single block of 256
    float f = ((float)i - 128.0f) * (1.0f / 512.0f);
    const float du = 6.0f / 4000.0f;
    float acc = 0.0f;
    for (int s = 0; s < 4001; ++s) {
        float u = -3.0f + du * (float)s;
        float k = kb_dev(u);
        acc += k * cosf(TWO_PI_F * f * u);
    }
    s1[i] = 1.0f / (acc * du);
}

// DFT matrix F[j,k] = exp(-2*pi*i*j*k/512), symmetric. Store real & imag planes.
__global__ void dftmat_kernel(float* __restrict__ Fr, float* __restrict__ Fi) {
    int i = blockIdx.x * blockDim.x + threadIdx.x;
    if (i >= G_ * G_) return;
    int j = i >> 9, k = i & 511;
    int m = (j * k) & 511;                     // jk mod 512 (exact in int)
    float ang = (float)m * (TWO_PI_F / 512.0f);
    Fr[i] = cosf(ang);
    Fi[i] = -sinf(ang);
}

// ---------------- per-frame kernels ----------------
// Apodize + zero-pad 256x256 -> 512x512 complex (separate r/i planes)
__global__ void pad_kernel(const float* __restrict__ x, const float* __restrict__ s1,
                           float* __restrict__ Pr, float* __restrict__ Pi, int t) {
    int i = blockIdx.x * blockDim.x + threadIdx.x;
    if (i >= G_ * G_) return;
    int j = i >> 9, k = i & 511;
    float vr = 0.0f, vi = 0.0f;
    if (j < H_ && k < W_) {
        float a = s1[j] * s1[k];
        int b = ((j * W_ + k) * 2) * T_ + t;   // x[0,0,j,k,c,t]
        vr = x[b] * a;
        vi = x[b + T_] * a;
    }
    Pr[i] = vr; Pi[i] = vi;
}

__global__ void zero2_kernel(float* __restrict__ a, float* __restrict__ b) {
    int i = blockIdx.x * blockDim.x + threadIdx.x;
    if (i >= G_ * G_) return;
    a[i] = 0.0f; b[i] = 0.0f;
}

// Complex GEMM C = (A * aSign_i) x (B * bSign_i), 512x512x512 row-major,
// fp32 WMMA 16x16x4. aSign/bSign scale only the imaginary planes (conj support).
// Block = 128 threads (4 waves) -> 32x32 C tile; B tiles (shared by all 4 waves)
// are staged into LDS with double-buffered GLOBAL_LOAD_ASYNC_TO_LDS_B128
// (ASYNCcnt) so the DMA of chunk k+1 overlaps the WMMA stream of chunk k.
__global__ __launch_bounds__(128)
void cgemm512_kernel(const float* __restrict__ Ar, const float* __restrict__ Ai,
                     const float* __restrict__ Br, const float* __restrict__ Bi,
                     float* __restrict__ Cr, float* __restrict__ Ci,
                     float aSign, float bSign, float scale) {
    const int N = G_;
    // double-buffered B tiles: [buf][KC_ rows][32 cols], real + imag planes
    __shared__ __align__(16) float tBr[2 * KC_ * 32];
    __shared__ __align__(16) float tBi[2 * KC_ * 32];

    int tid  = threadIdx.x;
    int lane = tid & 31;
    int wave = tid >> 5;                        // 0..3 -> 2x2 wave grid
    int blkN  = blockIdx.x * 32;
    int tileM = blockIdx.y * 32 + (wave >> 1) * 16;
    int tileN = blkN + (wave & 1) * 16;
    int half = lane >> 4;                       // K-half select for A/B frags
    int l15  = lane & 15;
    int rowA   = tileM + l15;                   // A-frag: lane carries row M
    int colLoc = (wave & 1) * 16 + l15;         // B col within the 32-wide tile

    // LDS byte-offset bases (low 32 bits of generic LDS address = LDS offset)
    uint32_t ldsBr = (uint32_t)(uintptr_t)tBr;
    uint32_t ldsBi = (uint32_t)(uintptr_t)tBi;
    const uint32_t bufStride = (uint32_t)(KC_ * 32 * 4);   // bytes per buffer

    // per-thread staging map: 4 x b128 per plane per chunk
    int chR[4], chC[4];
    #pragma unroll
    for (int p = 0; p < 4; ++p) {
        int chunk = tid + p * 128;              // 0..511
        chR[p] = chunk >> 3;                    // row 0..63
        chC[p] = (chunk & 7) << 2;              // col 0,4,..,28
    }

    v8f accRR = {0,0,0,0,0,0,0,0};
    v8f accII = {0,0,0,0,0,0,0,0};
    v8f accRI = {0,0,0,0,0,0,0,0};
    v8f accIR = {0,0,0,0,0,0,0,0};

    const float* arP = Ar + rowA * N;
    const float* aiP = Ai + rowA * N;

    // ---- preload chunk 0 into buffer 0 (async DMA to LDS) ----
    #pragma unroll
    for (int p = 0; p < 4; ++p) {
        uint32_t goff = (uint32_t)(((0 + chR[p]) * N + blkN + chC[p]) * 4);
        uint32_t loff = (uint32_t)((chR[p] * 32 + chC[p]) * 4);
        asm volatile("global_load_async_to_lds_b128 %0, %1, %2"
                     :: "v"(ldsBr + loff), "v"(goff), "s"(Br) : "memory");
        asm volatile("global_load_async_to_lds_b128 %0, %1, %2"
                     :: "v"(ldsBi + loff), "v"(goff), "s"(Bi) : "memory");
    }
    asm volatile("s_wait_asynccnt 0" ::: "memory");
    __syncthreads();

    for (int k0 = 0; k0 < N; k0 += KC_) {
        int cur = (k0 / KC_) & 1;
        uint32_t curOff = (uint32_t)cur * bufStride;
        // issue async DMA for the next chunk into the other buffer
        if (k0 + KC_ < N) {
            uint32_t nxtOff = (uint32_t)(1 - cur) * bufStride;
            #pragma unroll
            for (int p = 0; p < 4; ++p) {
                uint32_t goff = (uint32_t)(((k0 + KC_ + chR[p]) * N + blkN + chC[p]) * 4);
                uint32_t loff = nxtOff + (uint32_t)((chR[p] * 32 + chC[p]) * 4);
                asm volatile("global_load_async_to_lds_b128 %0, %1, %2"
                             :: "v"(ldsBr + loff), "v"(goff), "s"(Br) : "memory");
                asm volatile("global_load_async_to_lds_b128 %0, %1, %2"
                             :: "v"(ldsBi + loff), "v"(goff), "s"(Bi) : "memory");
            }
        }
        const float* lBr = tBr + cur * (KC_ * 32);
        const float* lBi = tBi + cur * (KC_ * 32);
        // ---- compute on current buffer ----
        #pragma unroll 4
        for (int kk = 0; kk < KC_; kk += 4) {
            int ka = kk + 2 * half;             // this lane's K pair in chunk
            // A-frag (16x4): contiguous pair from global (L2-resident)
            v2f ar = *(const v2f*)(arP + k0 + ka);
            v2f ai = *(const v2f*)(aiP + k0 + ka);
            ai.x *= aSign; ai.y *= aSign;
            __builtin_prefetch(arP + k0 + ka + KC_, 0, 3);
            // B-frag (4x16): rows ka, ka+1 of the LDS tile
            v2f br, bi;
            br.x = lBr[ka * 32 + colLoc];
            br.y = lBr[(ka + 1) * 32 + colLoc];
            bi.x = lBi[ka * 32 + colLoc] * bSign;
            bi.y = lBi[(ka + 1) * 32 + colLoc] * bSign;

            accRR = __builtin_amdgcn_wmma_f32_16x16x4_f32(false, ar, false, br,
                                                          (short)0, accRR, false, false);
            accII = __builtin_amdgcn_wmma_f32_16x16x4_f32(false, ai, false, bi,
                                                          (short)0, accII, false, false);
            accRI = __builtin_amdgcn_wmma_f32_16x16x4_f32(false, ar, false, bi,
                                                          (short)0, accRI, false, false);
            accIR = __builtin_amdgcn_wmma_f32_16x16x4_f32(false, ai, false, br,
                                                          (short)0, accIR, false, false);
        }
        // retire this wave's async DMA, then barrier: next buffer is ready and
        // everyone is done reading the buffer that iteration k0+KC_ refills.
        asm volatile("s_wait_asynccnt 0" ::: "memory");
        __syncthreads();
    }

    // D layout: VGPR r -> row tileM + r + 8*half, col tileN + l15
    #pragma unroll
    for (int r = 0; r < 8; ++r) {
        int row = tileM + r + 8 * half;
        int idx = row * N + tileN + l15;
        Cr[idx] = (accRR[r] - accII[r]) * scale;
        Ci[idx] = (accRI[r] + accIR[r]) * scale;
    }
}

// Fused: KB gather from Xg, DC-combine with y, dcf weight, KB scatter-adjoint.
__global__ void sample_kernel(const float* __restrict__ y, const float* __restrict__ ktraj,
                              const float* __restrict__ dcf, const float* __restrict__ lamp,
                              const float* __restrict__ Xr, const float* __restrict__ Xi,
                              float* __restrict__ Gr, float* __restrict__ Gi, int t) {
    int n = blockIdx.x * blockDim.x + threadIdx.x;
    if (n >= NRAD_) return;
    float lam = lamp[0];
    float kx = ktraj[n * T_ + t];                       // ktraj[0,0,n,t]
    float ky = ktraj[(NRAD_ + n) * T_ + t];             // ktraj[0,1,n,t]

    float tmx = fmodf(kx, TWO_PI_F); if (tmx < 0.0f) tmx += TWO_PI_F;
    float tmy = fmodf(ky, TWO_PI_F); if (tmy < 0.0f) tmy += TWO_PI_F;
    float gx = tmx * ((float)G_ / TWO_PI_F);
    float gy = tmy * ((float)G_ / TWO_PI_F);
    int bx = (int)floorf(gx);
    int by = (int)floorf(gy);

    float wx[JW], wy[JW];
    int ixv[JW], iyv[JW];
    #pragma unroll
    for (int j = 0; j < JW; ++j) {
        int ix = bx + j - 2;                            // offs = arange(6) - 2
        wx[j]  = kb_dev(gx - (float)ix);
        ixv[j] = (ix + G_) & (G_ - 1);
        int iy = by + j - 2;
        wy[j]  = kb_dev(gy - (float)iy);
        iyv[j] = (iy + G_) & (G_ - 1);
    }

    // gather: A_x = sum Xg[ix,iy] * wx*wy
    float axr = 0.0f, axi = 0.0f;
    #pragma unroll
    for (int jx = 0; jx < JW; ++jx) {
        int rowb = ixv[jx] * G_;
        float wxx = wx[jx];
        #pragma unroll
        for (int jy = 0; jy < JW; ++jy) {
            int f = rowb + iyv[jy];
            float w = wxx * wy[jy];
            axr += Xr[f] * w;
            axi += Xi[f] * w;
        }
    }

    float yr = y[(n * 2) * T_ + t];
    float yi = y[(n * 2 + 1) * T_ + t];
    float d  = dcf[n * T_ + t];
    float kr = d * (lam * axr + (1.0f - lam) * yr);
    float ki = d * (lam * axi + (1.0f - lam) * yi);

    // scatter-adjoint
    #pragma unroll
    for (int jx = 0; jx < JW; ++jx) {
        int rowb = ixv[jx] * G_;
        float wxx = wx[jx];
        #pragma unroll
        for (int jy = 0; jy < JW; ++jy) {
            int f = rowb + iyv[jy];
            float w = wxx * wy[jy];
            atomicAdd(&Gr[f], kr * w);
            atomicAdd(&Gi[f], ki * w);
        }
    }
}

// Crop 512x512 -> 256x256, apodize, write output (1,1,H,W,2,T)
__global__ void crop_kernel(const float* __restrict__ Rr, const float* __restrict__ Ri,
                            const float* __restrict__ s1, float* __restrict__ out, int t) {
    int i = blockIdx.x * blockDim.x + threadIdx.x;
    if (i >= H_ * W_) return;
    int h = i >> 8, w = i & 255;
    float a = s1[h] * s1[w];
    int g = h * G_ + w;
    int b = (i * 2) * T_ + t;
    out[b]      = Rr[g] * a;
    out[b + T_] = Ri[g] * a;
}

// ---------------- host launch ----------------
extern "C" void kernel_launch(void* const* d_in, const int* in_sizes, int n_in,
                              void* d_out, int out_size, void* d_ws, size_t ws_size,
                              hipStream_t stream) {
    const float* x     = (const float*)d_in[0];   // (1,1,256,256,2,24)
    const float* y     = (const float*)d_in[1];   // (1,1,24576,2,24)
    const float* ktraj = (const float*)d_in[2];   // (1,2,24576,24)
    const float* dcf   = (const float*)d_in[3];   // (1,1,1,24576,24)
    const float* lam   = (const float*)d_in[4];   // (1,)
    float* out = (float*)d_out;

    const size_t GG = (size_t)G_ * G_;
    float* ws = (float*)d_ws;
    float* s1 = ws;                 // 256
    float* Fr = s1 + 256;           // 512^2 each below
    float* Fi = Fr + GG;
    float* Pr = Fi + GG;  float* Pi = Pr + GG;
    float* Qr = Pi + GG;  float* Qi = Qr + GG;
    float* Rr = Qi + GG;  float* Ri = Rr + GG;

    const int TPB = 256;
    const int gridGG = (int)((GG + TPB - 1) / TPB);

    // setup: apodization vector + DFT matrix (deterministic each call)
    scale1d_kernel<<<1, 256, 0, stream>>>(s1);
    dftmat_kernel<<<gridGG, TPB, 0, stream>>>(Fr, Fi);

    dim3 gBlk(128);
    dim3 gGrd(G_ / 32, G_ / 32);    // 16x16 blocks of 32x32 tiles
    const float invG = 1.0f / (float)G_;   // ortho fwd 1/512; adjoint net 1/512

    for (int t = 0; t < T_; ++t) {
        // forward: Xg = F * (P * F) / 512
        pad_kernel<<<gridGG, TPB, 0, stream>>>(x, s1, Pr, Pi, t);
        cgemm512_kernel<<<gGrd, gBlk, 0, stream>>>(Pr, Pi, Fr, Fi, Qr, Qi,
                                                   1.0f, 1.0f, 1.0f);
        cgemm512_kernel<<<gGrd, gBlk, 0, stream>>>(Fr, Fi, Qr, Qi, Rr, Ri,
                                                   1.0f, 1.0f, invG);
        // gather + DC + dcf + scatter into zeroed P
        zero2_kernel<<<gridGG, TPB, 0, stream>>>(Pr, Pi);
        sample_kernel<<<NRAD_ / TPB, TPB, 0, stream>>>(y, ktraj, dcf, lam,
                                                       Rr, Ri, Pr, Pi, t);
        // adjoint: img = conj(F) * (P * conj(F)) / 512
        cgemm512_kernel<<<gGrd, gBlk, 0, stream>>>(Pr, Pi, Fr, Fi, Qr, Qi,
                                                   1.0f, -1.0f, 1.0f);
        cgemm512_kernel<<<gGrd, gBlk, 0, stream>>>(Fr, Fi, Qr, Qi, Rr, Ri,
                                                   -1.0f, 1.0f, invG);
        crop_kernel<<<(H_ * W_) / TPB, TPB, 0, stream>>>(Rr, Ri, s1, out, t);
    }
    (void)in_sizes; (void)n_in; (void)out_size; (void)ws_size;
}